// SheafMessagePassing_74285754352276
// MI455X (gfx1250) — compile-verified
//
#include <hip/hip_runtime.h>
#include <stdint.h>

// ---------------------------------------------------------------------------
// Sheaf message passing, 3 layers of:
//   S  = scatter_sum(x[src], dst)          (linearity: pulled W out of the sum)
//   x' = x @ W_self + S @ W_msg + b        (fp32 WMMA 16x16x4)
//   relu between layers
// ---------------------------------------------------------------------------

typedef float v2f __attribute__((ext_vector_type(2)));
typedef float v8f __attribute__((ext_vector_type(8)));

#define DF 64  // feature dim

__global__ __launch_bounds__(256) void zero_f32_kernel(float* __restrict__ p, int n) {
  int i = blockIdx.x * blockDim.x + threadIdx.x;
  if (i < n) p[i] = 0.0f;
}

// One wave32 per edge; lane l moves features [2l, 2l+1].
// x table (12.8 MB) and agg (12.8 MB) are L2-resident on MI455X (192 MB L2),
// so these atomics resolve in L2, never HBM.
__global__ __launch_bounds__(256) void sheaf_scatter_kernel(
    const float* __restrict__ x,
    const long long* __restrict__ src,
    const long long* __restrict__ dst,
    float* __restrict__ agg, int n_edges) {
  int tid = blockIdx.x * blockDim.x + threadIdx.x;
  int e    = tid >> 5;
  int lane = tid & 31;
  if (e >= n_edges) return;
  long long s = src[e];
  long long d = dst[e];
  v2f v = *(const v2f*)(x + s * DF + lane * 2);
  float* a = agg + d * DF + lane * 2;
  __hip_atomic_fetch_add(a,     v.x, __ATOMIC_RELAXED, __HIP_MEMORY_SCOPE_AGENT);
  __hip_atomic_fetch_add(a + 1, v.y, __ATOMIC_RELAXED, __HIP_MEMORY_SCOPE_AGENT);
}

// One wave per 16x16 output tile. Block = 128 threads = 4 waves = 4 column
// tiles (covers all 64 output columns). Grid.x = row tiles.
// fp32 WMMA per ISA 16x4 A layout: lanes 0-15 hold M rows, VGPR pair holds
// K={0,1} (half 0) or K={2,3} (half 1).
__global__ __launch_bounds__(128) void sheaf_gemm_kernel(
    const float* __restrict__ xin,   // [N,64] current features
    const float* __restrict__ agg,   // [N,64] scatter-summed neighbor features
    const float* __restrict__ wself, // [64,64]
    const float* __restrict__ wmsg,  // [64,64]
    const float* __restrict__ bias,  // [64]
    float* __restrict__ xout,        // [N,64]
    int n_rows, int relu) {
  const int wave = threadIdx.x >> 5;      // 0..3 -> column tile
  const int lane = threadIdx.x & 31;
  const int half = lane >> 4;             // K-pair selector
  const int r    = lane & 15;             // row (A) / column (B,C,D) within tile
  const int m    = blockIdx.x * 16;       // output row tile base
  const int n    = wave * 16;             // output col tile base

  int rowA = m + r;
  if (rowA >= n_rows) rowA = n_rows - 1;  // clamp loads; stores are guarded

  v8f c = {};

  // x @ W_self
  {
    const float* arow = xin + (long long)rowA * DF;
    const float* bcol = wself + n + r;
#pragma unroll
    for (int k0 = 0; k0 < DF; k0 += 4) {
      v2f a = *(const v2f*)(arow + k0 + 2 * half);
      v2f b;
      b.x = bcol[(k0 + 2 * half) * DF];
      b.y = bcol[(k0 + 2 * half + 1) * DF];
      c = __builtin_amdgcn_wmma_f32_16x16x4_f32(
          /*neg_a=*/false, a, /*neg_b=*/false, b,
          /*c_mod=*/(short)0, c, /*reuse_a=*/false, /*reuse_b=*/false);
    }
  }

  // S @ W_msg accumulated into the same tile
  {
    const float* arow = agg + (long long)rowA * DF;
    const float* bcol = wmsg + n + r;
#pragma unroll
    for (int k0 = 0; k0 < DF; k0 += 4) {
      v2f a = *(const v2f*)(arow + k0 + 2 * half);
      v2f b;
      b.x = bcol[(k0 + 2 * half) * DF];
      b.y = bcol[(k0 + 2 * half + 1) * DF];
      c = __builtin_amdgcn_wmma_f32_16x16x4_f32(
          false, a, false, b, (short)0, c, false, false);
    }
  }

  // bias + relu + store.  C/D layout: VGPR j, lanes 0-15 -> M=j, lanes 16-31 -> M=j+8.
  const float bb = bias[n + r];
#pragma unroll
  for (int j = 0; j < 8; ++j) {
    int row = m + j + 8 * half;
    if (row < n_rows) {
      float v = c[j] + bb;
      if (relu) v = fmaxf(v, 0.0f);
      xout[(long long)row * DF + (n + r)] = v;
    }
  }
}

extern "C" void kernel_launch(void* const* d_in, const int* in_sizes, int n_in,
                              void* d_out, int out_size, void* d_ws, size_t ws_size,
                              hipStream_t stream) {
  // setup_inputs order: x[N,64] f32, edge_index[2,E] i64, W_msg[L,64,64] f32,
  //                     W_self[L,64,64] f32, b[L,64] f32
  const float*     x0    = (const float*)d_in[0];
  const long long* ei    = (const long long*)d_in[1];
  const float*     Wmsg  = (const float*)d_in[2];
  const float*     Wself = (const float*)d_in[3];
  const float*     bvec  = (const float*)d_in[4];
  float*           out   = (float*)d_out;

  const int N = in_sizes[0] / DF;              // 50000
  const int E = in_sizes[1] / 2;               // 800000
  const int n_layers = in_sizes[4] / DF;       // 3

  const long long* src = ei;
  const long long* dst = ei + E;

  float* bufA = (float*)d_ws;                  // [N,64]
  float* bufB = bufA + (size_t)N * DF;         // [N,64]
  float* agg  = bufB + (size_t)N * DF;         // [N,64]
  const int nd = N * DF;

  const int row_tiles = (N + 15) / 16;

  const float* cur = x0;
  for (int layer = 0; layer < n_layers; ++layer) {
    zero_f32_kernel<<<(nd + 255) / 256, 256, 0, stream>>>(agg, nd);

    // 32 lanes per edge
    long long scatter_threads = (long long)E * 32;
    int scatter_blocks = (int)((scatter_threads + 255) / 256);
    sheaf_scatter_kernel<<<scatter_blocks, 256, 0, stream>>>(cur, src, dst, agg, E);

    float* nxt = (layer == n_layers - 1) ? out : ((layer & 1) ? bufB : bufA);
    sheaf_gemm_kernel<<<row_tiles, 128, 0, stream>>>(
        cur, agg,
        Wself + (size_t)layer * DF * DF,
        Wmsg  + (size_t)layer * DF * DF,
        bvec  + (size_t)layer * DF,
        nxt, N, (layer < n_layers - 1) ? 1 : 0);
    cur = nxt;
  }
}